// PaGNN_56255481643193
// MI455X (gfx1250) — compile-verified
//
#include <hip/hip_runtime.h>

typedef __attribute__((ext_vector_type(2))) float v2f;
typedef __attribute__((ext_vector_type(8))) float v8f;

#define D_FEAT 128
#define HIDDEN 128
#define NCLS   40

// ---------------- zero workspace (accumulators) ----------------
__global__ __launch_bounds__(256) void k_zero(float* __restrict__ p, long n) {
  long i = (long)blockIdx.x * blockDim.x + threadIdx.x;
  long stride = (long)gridDim.x * blockDim.x;
  for (; i < n; i += stride) p[i] = 0.0f;
}

// ---------------- edge scatter 1: agg / per-feature deg / scalar deg ----------------
// one wave (32 lanes) per edge; each lane handles 4 features (float4 / uchar4)
__global__ __launch_bounds__(256) void k_scatter1(
    const float* __restrict__ x, const unsigned char* __restrict__ mask,
    const int* __restrict__ ei, float* __restrict__ agg, float* __restrict__ deg,
    float* __restrict__ degS, int nE) {
  long gid = (long)blockIdx.x * blockDim.x + threadIdx.x;
  long e = gid >> 5;
  int lane = (int)(gid & 31);
  if (e >= nE) return;
  int r = ei[e];
  int c = ei[(long)nE + e];
  long coff = (long)c * D_FEAT + lane * 4;
  float4 xv = *(const float4*)(x + coff);
  uchar4 mv = *(const uchar4*)(mask + coff);
  float m0 = mv.x ? 1.0f : 0.0f;
  float m1 = mv.y ? 1.0f : 0.0f;
  float m2 = mv.z ? 1.0f : 0.0f;
  float m3 = mv.w ? 1.0f : 0.0f;
  long roff = (long)r * D_FEAT + lane * 4;
  atomicAdd(agg + roff + 0, xv.x * m0);
  atomicAdd(agg + roff + 1, xv.y * m1);
  atomicAdd(agg + roff + 2, xv.z * m2);
  atomicAdd(agg + roff + 3, xv.w * m3);
  atomicAdd(deg + roff + 0, m0);
  atomicAdd(deg + roff + 1, m1);
  atomicAdd(deg + roff + 2, m2);
  atomicAdd(deg + roff + 3, m3);
  if (lane == 0) atomicAdd(degS + r, 1.0f);
}

// ---------------- normalize: agg = agg / clip(deg, 1)  (in place, float4) ----------------
__global__ __launch_bounds__(256) void k_normalize(
    float* __restrict__ agg, const float* __restrict__ deg, long n4) {
  long i = (long)blockIdx.x * blockDim.x + threadIdx.x;
  if (i >= n4) return;
  float4 a = ((const float4*)agg)[i];
  float4 d = ((const float4*)deg)[i];
  a.x = a.x / fmaxf(d.x, 1.0f);
  a.y = a.y / fmaxf(d.y, 1.0f);
  a.z = a.z / fmaxf(d.z, 1.0f);
  a.w = a.w / fmaxf(d.w, 1.0f);
  ((float4*)agg)[i] = a;
}

// ---------------- dinv = rsqrt(edge_count + 1 self loop) ----------------
__global__ __launch_bounds__(256) void k_dinv(float* __restrict__ degS, int n) {
  int i = blockIdx.x * blockDim.x + threadIdx.x;
  if (i < n) degS[i] = rsqrtf(degS[i] + 1.0f);  // always >= 1 -> clip is a no-op
}

// ---------------- GEMM1: h = relu(hin @ W1 + b1), in-place into hin ----------------
// 256 threads = 8 waves; each wave owns a 16-row tile.
// W1 staged in LDS k-pair-swizzled: lds[((k>>1)*128 + n)*2 + (k&1)] = W1[k][n],
// so a B fragment {W1[k][n], W1[k+1][n]} (k even) is one aligned ds_load_b64.
__global__ __launch_bounds__(256) void k_gemm1(
    const float* __restrict__ hin, const float* __restrict__ W1,
    const float* __restrict__ b1, float* __restrict__ h, int n) {
  __shared__ float w1s[HIDDEN * HIDDEN];  // 64 KB
  int t = threadIdx.x;
  {
#pragma unroll
    for (int i = 0; i < 16; ++i) {
      int g = t + i * 256;              // float4 index, 4096 total
      int r = g >> 5;                   // row (128 floats = 32 float4 per row)
      int c = (g & 31) * 4;             // starting col
      float4 v = ((const float4*)W1)[g];
      int base = ((r >> 1) * HIDDEN) * 2 + (r & 1);
      w1s[base + (c + 0) * 2] = v.x;
      w1s[base + (c + 1) * 2] = v.y;
      w1s[base + (c + 2) * 2] = v.z;
      w1s[base + (c + 3) * 2] = v.w;
    }
  }
  __syncthreads();

  int wave = t >> 5, lane = t & 31;
  int half = lane >> 4, m16 = lane & 15;
  long row0 = ((long)blockIdx.x * 8 + wave) * 16;
  if (row0 >= n) return;

  v8f zero8 = {0.f, 0.f, 0.f, 0.f, 0.f, 0.f, 0.f, 0.f};
  v8f acc[8];
#pragma unroll
  for (int ct = 0; ct < 8; ++ct) acc[ct] = zero8;

  long arow = row0 + m16;
  bool rvalid = (arow < n);
  long crow = rvalid ? arow : (long)(n - 1);   // clamp: load always legal, no EXEC divergence
  const float* ap = hin + crow * (long)D_FEAT;
  const v2f* w1v = (const v2f*)w1s;

  for (int ks = 0; ks < 32; ++ks) {        // K = 128 in steps of 4
    int k = ks * 4 + half * 2;             // lanes 0-15: K+{0,1}; lanes 16-31: K+{2,3}
    int pair = ks * 2 + half;              // k >> 1
    float2 a = *(const float2*)(ap + k);
    v2f av;
    av.x = rvalid ? a.x : 0.0f;            // branchless select, EXEC stays all-1s
    av.y = rvalid ? a.y : 0.0f;
#pragma unroll
    for (int ct = 0; ct < 8; ++ct) {
      int ncol = ct * 16 + m16;
      v2f bv = w1v[pair * HIDDEN + ncol];  // single ds_load_b64, no repack movs
      acc[ct] = __builtin_amdgcn_wmma_f32_16x16x4_f32(
          false, av, false, bv, (short)0, acc[ct], false, false);
    }
  }
  // epilogue: + b1, relu, store (D layout: acc[ct][j] = D[j + 8*half][m16])
#pragma unroll
  for (int ct = 0; ct < 8; ++ct) {
    int ncol = ct * 16 + m16;
    float bias = b1[ncol];
#pragma unroll
    for (int j = 0; j < 8; ++j) {
      long r = row0 + j + 8 * half;
      if (r < n) h[r * (long)HIDDEN + ncol] = fmaxf(acc[ct][j] + bias, 0.0f);
    }
  }
}

// ---------------- GEMM2: xw = h @ W2  (W2 zero-padded to 128x48, k-pair-swizzled LDS) ----
__global__ __launch_bounds__(256) void k_gemm2(
    const float* __restrict__ h, const float* __restrict__ W2,
    float* __restrict__ xw, int n) {
  __shared__ float w2s[HIDDEN * 48];  // 24 KB
  int t = threadIdx.x;
  for (int i = t; i < HIDDEN * 48; i += 256) {
    int rr = i / 48, cc = i - rr * 48;
    float v = (cc < NCLS) ? W2[rr * NCLS + cc] : 0.0f;
    w2s[((rr >> 1) * 48 + cc) * 2 + (rr & 1)] = v;
  }
  __syncthreads();

  int wave = t >> 5, lane = t & 31;
  int half = lane >> 4, m16 = lane & 15;
  long row0 = ((long)blockIdx.x * 8 + wave) * 16;
  if (row0 >= n) return;

  v8f zero8 = {0.f, 0.f, 0.f, 0.f, 0.f, 0.f, 0.f, 0.f};
  v8f acc[3];
#pragma unroll
  for (int ct = 0; ct < 3; ++ct) acc[ct] = zero8;

  long arow = row0 + m16;
  bool rvalid = (arow < n);
  long crow = rvalid ? arow : (long)(n - 1);
  const float* ap = h + crow * (long)HIDDEN;
  const v2f* w2v = (const v2f*)w2s;

  for (int ks = 0; ks < 32; ++ks) {
    int k = ks * 4 + half * 2;
    int pair = ks * 2 + half;
    float2 a = *(const float2*)(ap + k);
    v2f av;
    av.x = rvalid ? a.x : 0.0f;
    av.y = rvalid ? a.y : 0.0f;
#pragma unroll
    for (int ct = 0; ct < 3; ++ct) {
      int ncol = ct * 16 + m16;
      v2f bv = w2v[pair * 48 + ncol];
      acc[ct] = __builtin_amdgcn_wmma_f32_16x16x4_f32(
          false, av, false, bv, (short)0, acc[ct], false, false);
    }
  }
#pragma unroll
  for (int ct = 0; ct < 3; ++ct) {
    int ncol = ct * 16 + m16;
    if (ncol < NCLS) {
#pragma unroll
      for (int j = 0; j < 8; ++j) {
        long r = row0 + j + 8 * half;
        if (r < n) xw[r * (long)NCLS + ncol] = acc[ct][j];
      }
    }
  }
}

// ---------------- out init: self-loop contribution dinv[i]^2 * xw[i] ----------------
__global__ __launch_bounds__(256) void k_outinit(
    float* __restrict__ out, const float* __restrict__ xw,
    const float* __restrict__ dinv, int n) {
  long gid = (long)blockIdx.x * blockDim.x + threadIdx.x;
  if (gid >= (long)n * NCLS) return;
  int i = (int)(gid / NCLS);
  float di = dinv[i];
  out[gid] = di * di * xw[gid];
}

// ---------------- edge scatter 2: out[r] += dinv[r]*dinv[c]*xw[c] ----------------
__global__ __launch_bounds__(256) void k_scatter2(
    const int* __restrict__ ei, const float* __restrict__ dinv,
    const float* __restrict__ xw, float* __restrict__ out, int nE) {
  long gid = (long)blockIdx.x * blockDim.x + threadIdx.x;
  if (gid >= (long)nE * NCLS) return;
  long e = gid / NCLS;
  int f = (int)(gid - e * NCLS);
  int r = ei[e];
  int c = ei[(long)nE + e];
  float norm = dinv[r] * dinv[c];
  atomicAdd(out + (long)r * NCLS + f, norm * xw[(long)c * NCLS + f]);
}

// ---------------- log_softmax over 40 classes: one wave32 per node ----------------
__global__ __launch_bounds__(256) void k_logsoftmax(
    float* __restrict__ out, const float* __restrict__ b2, int n) {
  long gid = (long)blockIdx.x * blockDim.x + threadIdx.x;
  long node = gid >> 5;
  int lane = (int)(gid & 31);
  if (node >= n) return;
  float* op = out + node * (long)NCLS;
  int f2 = lane + 32;
  float a = op[lane] + b2[lane];                                   // lane < 32 <= 40
  float b = (f2 < NCLS) ? op[f2] + b2[f2] : -__builtin_inff();
  float m = fmaxf(a, b);
#pragma unroll
  for (int off = 16; off > 0; off >>= 1) m = fmaxf(m, __shfl_xor(m, off, 32));
  float s = expf(a - m) + ((f2 < NCLS) ? expf(b - m) : 0.0f);
#pragma unroll
  for (int off = 16; off > 0; off >>= 1) s += __shfl_xor(s, off, 32);
  float lse = m + logf(s);
  op[lane] = a - lse;
  if (f2 < NCLS) op[f2] = b - lse;
}

extern "C" void kernel_launch(void* const* d_in, const int* in_sizes, int n_in,
                              void* d_out, int out_size, void* d_ws, size_t ws_size,
                              hipStream_t stream) {
  const float* x          = (const float*)d_in[0];
  const unsigned char* mk = (const unsigned char*)d_in[1];
  const int* ei           = (const int*)d_in[2];
  const float* W1         = (const float*)d_in[3];
  const float* b1         = (const float*)d_in[4];
  const float* W2         = (const float*)d_in[5];
  const float* b2         = (const float*)d_in[6];
  float* out              = (float*)d_out;

  int N = in_sizes[0] / D_FEAT;
  int E = in_sizes[2] / 2;

  // workspace layout (floats): agg[N*128] (-> hin -> h) | deg[N*128] | degS[N] (-> dinv) | xw[N*40]
  float* agg  = (float*)d_ws;
  float* deg  = agg + (long)N * D_FEAT;
  float* degS = deg + (long)N * D_FEAT;
  float* xw   = degS + N;

  // 1) zero accumulators (agg + deg + degS are contiguous)
  k_zero<<<2048, 256, 0, stream>>>(agg, (long)N * (2 * D_FEAT + 1));

  // 2) PaGNN aggregation over edges
  long s1threads = (long)E * 32;
  k_scatter1<<<(int)((s1threads + 255) / 256), 256, 0, stream>>>(x, mk, ei, agg, deg, degS, E);

  // 3) hin = agg / clip(deg, 1)  (in place, exact IEEE div, hoisted out of GEMM)
  long n4 = (long)N * (D_FEAT / 4);
  k_normalize<<<(int)((n4 + 255) / 256), 256, 0, stream>>>(agg, deg, n4);

  // 4) dinv for GCN (self-loop included)
  k_dinv<<<(N + 255) / 256, 256, 0, stream>>>(degS, N);

  // 5) h = relu(hin @ W1 + b1)   [WMMA f32, in-place into agg]
  int gblocks = (N + 127) / 128;
  k_gemm1<<<gblocks, 256, 0, stream>>>(agg, W1, b1, agg, N);

  // 6) xw = h @ W2   [WMMA f32]
  k_gemm2<<<gblocks, 256, 0, stream>>>(agg, W2, xw, N);

  // 7) out = dinv^2 * xw (self loops)
  long oithreads = (long)N * NCLS;
  k_outinit<<<(int)((oithreads + 255) / 256), 256, 0, stream>>>(out, xw, degS, N);

  // 8) out[r] += dinv[r]*dinv[c]*xw[c] over edges
  long s2threads = (long)E * NCLS;
  k_scatter2<<<(int)((s2threads + 255) / 256), 256, 0, stream>>>(ei, degS, xw, out, E);

  // 9) out = log_softmax(out + b2)
  long lsthreads = (long)N * 32;
  k_logsoftmax<<<(int)((lsthreads + 255) / 256), 256, 0, stream>>>(out, b2, N);
}